// MeshSplatter_88837103551025
// MI455X (gfx1250) — compile-verified
//
#include <hip/hip_runtime.h>
#include <math.h>

#define NPTS  1024
#define IMG_H 192
#define IMG_W 256

typedef float v2f __attribute__((ext_vector_type(2)));
typedef float v8f __attribute__((ext_vector_type(8)));
typedef unsigned int u32x4 __attribute__((ext_vector_type(4)));
typedef int i32x8 __attribute__((ext_vector_type(8)));
typedef int i32x4 __attribute__((ext_vector_type(4)));

__device__ __forceinline__ float clampf(float x, float lo, float hi) {
    return fminf(fmaxf(x, lo), hi);
}

// ---------------------------------------------------------------------------
// Phase 1: per-gaussian preprocessing (projection, conic, SH color)
// ---------------------------------------------------------------------------
__global__ void __launch_bounds__(256)
preprocess_kernel(const float* __restrict__ points,
                  const float* __restrict__ mean_params,
                  const float* __restrict__ scale_params,
                  const float* __restrict__ quat_params,
                  const float* __restrict__ colors,
                  const float* __restrict__ shs,
                  const float* __restrict__ vm,   // 4x4 row-major
                  const float* __restrict__ Km,   // 3x3 row-major
                  float*  __restrict__ zkey,
                  float4* __restrict__ F0,        // (u, v, cA, cB)
                  float*  __restrict__ CC,        // cC
                  float4* __restrict__ RGBV)      // (r, g, b, valid)
{
    const int n = blockIdx.x * blockDim.x + threadIdx.x;
    if (n >= NPTS) return;

    const float C0c = 0.28209479177387814f;
    const float C1c = 0.4886025119029199f;
    const float C2_0 = 1.0925484305920792f, C2_1 = -1.0925484305920792f,
                C2_2 = 0.31539156525252005f, C2_3 = -1.0925484305920792f,
                C2_4 = 0.5462742152960396f;
    const float C3_0 = -0.5900435899266435f, C3_1 = 2.890611442640554f,
                C3_2 = -0.4570457994644658f, C3_3 = 0.3731763325901154f,
                C3_4 = -0.4570457994644658f, C3_5 = 1.445305721320277f,
                C3_6 = -0.5900435899266435f;
    const float LOG_MAX_SCALE = -1.8325814637483102f; // ln(0.16)

    // means
    float mean[3];
    #pragma unroll
    for (int i = 0; i < 3; ++i) {
        float mp = clampf(mean_params[n * 3 + i], -0.05f, 0.05f);
        mean[i] = points[n * 3 + i] + mp;
    }

    // scales
    float l0 = clampf(scale_params[n * 2 + 0], -7.0f, LOG_MAX_SCALE);
    float l1 = clampf(scale_params[n * 2 + 1], -7.0f, LOG_MAX_SCALE);
    float s0 = __expf(l0), s1 = __expf(l1), s2 = __expf(-7.0f);
    float sq[3] = { s0 * s0, s1 * s1, s2 * s2 };

    // quaternion -> rotation
    float qx = quat_params[n * 4 + 0], qy = quat_params[n * 4 + 1];
    float qz = quat_params[n * 4 + 2], qw = quat_params[n * 4 + 3];
    float qinv = rsqrtf(qx * qx + qy * qy + qz * qz + qw * qw);
    qx *= qinv; qy *= qinv; qz *= qinv; qw *= qinv;
    float R[3][3];
    R[0][0] = 1.f - 2.f * (qy * qy + qz * qz);
    R[0][1] = 2.f * (qx * qy - qw * qz);
    R[0][2] = 2.f * (qx * qz + qw * qy);
    R[1][0] = 2.f * (qx * qy + qw * qz);
    R[1][1] = 1.f - 2.f * (qx * qx + qz * qz);
    R[1][2] = 2.f * (qy * qz - qw * qx);
    R[2][0] = 2.f * (qx * qz - qw * qy);
    R[2][1] = 2.f * (qy * qz + qw * qx);
    R[2][2] = 1.f - 2.f * (qx * qx + qy * qy);

    float cov3d[3][3];
    #pragma unroll
    for (int i = 0; i < 3; ++i)
        #pragma unroll
        for (int j = 0; j < 3; ++j)
            cov3d[i][j] = R[i][0] * sq[0] * R[j][0] +
                          R[i][1] * sq[1] * R[j][1] +
                          R[i][2] * sq[2] * R[j][2];

    // camera transform
    float Wr[3][3], tvec[3];
    #pragma unroll
    for (int i = 0; i < 3; ++i) {
        #pragma unroll
        for (int j = 0; j < 3; ++j) Wr[i][j] = vm[i * 4 + j];
        tvec[i] = vm[i * 4 + 3];
    }
    float pc[3];
    #pragma unroll
    for (int i = 0; i < 3; ++i)
        pc[i] = Wr[i][0] * mean[0] + Wr[i][1] * mean[1] + Wr[i][2] * mean[2] + tvec[i];
    const float x = pc[0], y = pc[1], z = pc[2];

    const float fx = Km[0], fy = Km[4], cx = Km[2], cy = Km[5];
    const float NEAR = 0.01f, BLUR = 0.3f;

    float zc = (z > NEAR) ? z : 1.0f;
    float rz = 1.0f / zc;
    float limx = 1.3f * (0.5f * (float)IMG_W / fx);
    float limy = 1.3f * (0.5f * (float)IMG_H / fy);
    float tx = zc * clampf(x * rz, -limx, limx);
    float ty = zc * clampf(y * rz, -limy, limy);
    float rz2 = rz * rz;
    float J00 = fx * rz, J02 = -fx * tx * rz2;
    float J11 = fy * rz, J12 = -fy * ty * rz2;

    float M0[3], M1[3];
    #pragma unroll
    for (int j = 0; j < 3; ++j) {
        M0[j] = J00 * Wr[0][j] + J02 * Wr[2][j];
        M1[j] = J11 * Wr[1][j] + J12 * Wr[2][j];
    }
    float t0[3], t1[3];
    #pragma unroll
    for (int k = 0; k < 3; ++k) {
        t0[k] = M0[0] * cov3d[0][k] + M0[1] * cov3d[1][k] + M0[2] * cov3d[2][k];
        t1[k] = M1[0] * cov3d[0][k] + M1[1] * cov3d[1][k] + M1[2] * cov3d[2][k];
    }
    float a = t0[0] * M0[0] + t0[1] * M0[1] + t0[2] * M0[2] + BLUR;
    float b = t0[0] * M1[0] + t0[1] * M1[1] + t0[2] * M1[2];
    float c = t1[0] * M1[0] + t1[1] * M1[1] + t1[2] * M1[2] + BLUR;
    float det = a * c - b * b;
    bool valid = (z > NEAR) && (det > 1e-12f);
    float dets = (det > 1e-12f) ? det : 1.0f;
    float dinv = 1.0f / dets;
    float cA = c * dinv, cB = -b * dinv, cCv = a * dinv;

    float u = fx * x * rz + cx;
    float v = fy * y * rz + cy;

    // view direction
    float campos[3];
    #pragma unroll
    for (int i = 0; i < 3; ++i)
        campos[i] = -(Wr[0][i] * tvec[0] + Wr[1][i] * tvec[1] + Wr[2][i] * tvec[2]);
    float dx = mean[0] - campos[0], dy = mean[1] - campos[1], dz = mean[2] - campos[2];
    float dn = sqrtf(dx * dx + dy * dy + dz * dz);
    dn = fmaxf(dn, 1e-8f);
    dx /= dn; dy /= dn; dz /= dn;

    // SH basis (deg 3)
    float xx = dx * dx, yy = dy * dy, zz = dz * dz;
    float xy = dx * dy, yz = dy * dz, xz = dx * dz;
    float bas[16];
    bas[0] = C0c;
    bas[1] = -C1c * dy;  bas[2] = C1c * dz;  bas[3] = -C1c * dx;
    bas[4] = C2_0 * xy;  bas[5] = C2_1 * yz;
    bas[6] = C2_2 * (2.f * zz - xx - yy);
    bas[7] = C2_3 * xz;  bas[8] = C2_4 * (xx - yy);
    bas[9]  = C3_0 * dy * (3.f * xx - yy);
    bas[10] = C3_1 * xy * dz;
    bas[11] = C3_2 * dy * (4.f * zz - xx - yy);
    bas[12] = C3_3 * dz * (2.f * zz - 3.f * xx - 3.f * yy);
    bas[13] = C3_4 * dx * (4.f * zz - xx - yy);
    bas[14] = C3_5 * dz * (xx - yy);
    bas[15] = C3_6 * dx * (xx - 3.f * yy);

    float rgb[3];
    #pragma unroll
    for (int ch = 0; ch < 3; ++ch) {
        float sh0 = (colors[n * 3 + ch] - 0.5f) / C0c;
        float acc = bas[0] * sh0;
        #pragma unroll
        for (int k = 1; k < 16; ++k)
            acc += bas[k] * shs[n * 45 + (k - 1) * 3 + ch];
        rgb[ch] = fmaxf(acc + 0.5f, 0.0f);
    }

    zkey[n] = z;
    F0[n]   = make_float4(u, v, cA, cB);
    CC[n]   = cCv;
    RGBV[n] = make_float4(rgb[0], rgb[1], rgb[2], valid ? 1.0f : 0.0f);
}

// ---------------------------------------------------------------------------
// Phase 2: stable depth sort (bitonic on (z, idx)) + gather
// ---------------------------------------------------------------------------
__global__ void __launch_bounds__(1024)
sort_kernel(const float*  __restrict__ zkey,
            const float4* __restrict__ F0,
            const float*  __restrict__ CC,
            const float4* __restrict__ RGBV,
            float4* __restrict__ SF0,
            float*  __restrict__ SCC,
            float4* __restrict__ SRGBV)
{
    __shared__ float skey[NPTS];
    __shared__ int   sidx[NPTS];
    const int t = threadIdx.x;
    skey[t] = zkey[t];
    sidx[t] = t;
    __syncthreads();

    for (int k = 2; k <= NPTS; k <<= 1) {
        for (int j = k >> 1; j > 0; j >>= 1) {
            int ixj = t ^ j;
            if (ixj > t) {
                float av = skey[t], bv = skey[ixj];
                int   ai = sidx[t], bi = sidx[ixj];
                bool aGreater = (av > bv) || (av == bv && ai > bi);
                bool up = ((t & k) == 0);
                if (aGreater == up) {
                    skey[t] = bv; skey[ixj] = av;
                    sidx[t] = bi; sidx[ixj] = ai;
                }
            }
            __syncthreads();
        }
    }
    int s = sidx[t];
    SF0[t]   = F0[s];
    SCC[t]   = CC[s];
    SRGBV[t] = RGBV[s];
}

// ---------------------------------------------------------------------------
// TDM: 1-D DMA copy of `ndw` dwords from global to LDS.
// D# per CDNA5 ISA ch.8: group0 = count/lds_addr/global_addr/type,
// group1 = data_size=4B, tensor_dim0=tile_dim0=ndw, dim1=1, stride=ndw.
// Trailing groups zero (tile_dim2 = 0 -> unused).
// This toolchain exposes the 6-arg builtin:
//   (u32x4 g0, i32x8 g1, i32x4, i32x4, i32x8, i32 cpol)
// ---------------------------------------------------------------------------
#if defined(__has_builtin)
#if __has_builtin(__builtin_amdgcn_tensor_load_to_lds)
#define HAVE_TDM 1
#endif
#endif

__device__ __forceinline__ void tdm_load_1d(unsigned lds_addr, const void* gptr,
                                            unsigned ndw) {
#ifdef HAVE_TDM
    unsigned long long ga = (unsigned long long)(uintptr_t)gptr;
    u32x4 g0;
    g0.x = 0x1u;                                           // count=1, user mode
    g0.y = lds_addr;                                       // lds_addr [63:32]
    g0.z = (unsigned)(ga & 0xffffffffu);                   // global_addr lo
    g0.w = (unsigned)((ga >> 32) & 0x1ffffffu) | (2u << 30); // addr hi | type=2
    i32x8 g1;
    g1[0] = 0x00020000;                       // wg_mask=0, data_size=2 (4B)
    g1[1] = (int)((ndw & 0xffffu) << 16);     // tensor_dim0 lo16 @ [31:16]
    g1[2] = (int)((ndw >> 16) & 0xffffu) | (1 << 16); // dim0 hi16 | tensor_dim1=1 lo16
    g1[3] = (int)((ndw & 0xffffu) << 16);     // tensor_dim1 hi=0 | tile_dim0=ndw
    g1[4] = 1;                                // tile_dim1=1, tile_dim2=0 (unused)
    g1[5] = (int)ndw;                         // tensor_dim0_stride lo32
    g1[6] = 0;                                // stride0 hi16 | stride1 lo16
    g1[7] = 0;                                // stride1 hi32
    i32x4 z4 = { 0, 0, 0, 0 };
    i32x8 z8 = { 0, 0, 0, 0, 0, 0, 0, 0 };
    __builtin_amdgcn_tensor_load_to_lds(g0, g1, z4, z4, z8, 0);
#else
    (void)lds_addr; (void)gptr; (void)ndw;
#endif
}

// ---------------------------------------------------------------------------
// Phase 3: tiled rasterizer. One wave32 = 32 consecutive pixels of one row.
// sigma via V_WMMA_F32_16X16X4_F32: A = per-gaussian (k0,k1,k2,0),
// B = per-pixel (X^2, X, 1, 0) in tile-local coords.
// ---------------------------------------------------------------------------
__global__ void __launch_bounds__(256)
raster_kernel(const float4* __restrict__ SF0,
              const float*  __restrict__ SCC,
              const float4* __restrict__ SRGBV,
              float* __restrict__ out)
{
    __shared__ float4 sF0[NPTS];    // 16 KB
    __shared__ float  sCC[NPTS];    //  4 KB
    __shared__ float4 sRGBV[NPTS];  // 16 KB

    const int t = threadIdx.x;
#ifdef HAVE_TDM
    if (t < 32) {   // wave 0 issues the three DMA descriptors (per-wave op)
        tdm_load_1d((unsigned)(uintptr_t)(void*)sF0,   SF0,   NPTS * 4);
        tdm_load_1d((unsigned)(uintptr_t)(void*)sCC,   SCC,   NPTS);
        tdm_load_1d((unsigned)(uintptr_t)(void*)sRGBV, SRGBV, NPTS * 4);
        __builtin_amdgcn_s_wait_tensorcnt(0);
    }
#else
    __builtin_prefetch(SF0, 0, 3);
    __builtin_prefetch(SRGBV, 0, 3);
    for (int i = t; i < NPTS; i += 256) {
        sF0[i]   = SF0[i];
        sCC[i]   = SCC[i];
        sRGBV[i] = SRGBV[i];
    }
#endif
    __syncthreads();

    const int wave = t >> 5;
    const int lane = t & 31;
    const int tile = blockIdx.x * 8 + wave;          // 1536 tiles total
    const int row  = tile >> 3;                      // tiles_per_row = 8
    const int tx0  = (tile & 7) << 5;                // 32 pixels per tile
    const float py = (float)row + 0.5f;

    const int p = lane & 15;   // column within 16x16 WMMA / gaussian slot
    const int h = lane >> 4;   // K-half selector

    // B operands: constant per tile (pixel basis, tile-local X)
    const float Xlo = (float)p + 0.5f;
    const float Xhi = (float)p + 16.5f;
    v2f Blo, Bhi;
    Blo.x = h ? 1.0f : Xlo * Xlo;  Blo.y = h ? 0.0f : Xlo;
    Bhi.x = h ? 1.0f : Xhi * Xhi;  Bhi.y = h ? 0.0f : Xhi;

    const v8f zeroC = { 0.f, 0.f, 0.f, 0.f, 0.f, 0.f, 0.f, 0.f };

    float T = 1.0f;
    float accR = 0.f, accG = 0.f, accB = 0.f;
    const bool lo = (lane < 16);

    for (int cb = 0; cb < NPTS; cb += 16) {
        // build A: per-gaussian sigma polynomial coefficients
        const int g = cb + p;
        float4 f0 = sF0[g];
        float cC  = sCC[g];
        float U   = f0.x - (float)tx0;     // gaussian center, tile-local
        float DY  = py - f0.y;
        float cA  = f0.z, cBc = f0.w;
        float k0 = 0.5f * cA;
        float k1 = cBc * DY - cA * U;
        float k2 = (0.5f * cA * U - cBc * DY) * U + 0.5f * cC * DY * DY;
        v2f A;
        A.x = h ? k2 : k0;
        A.y = h ? 0.0f : k1;

        // sigma = A(16x4) x B(4x16): gaussians x pixels
        v8f d0 = __builtin_amdgcn_wmma_f32_16x16x4_f32(
            false, A, false, Blo, (short)0, zeroC, false, false);
        v8f d1 = __builtin_amdgcn_wmma_f32_16x16x4_f32(
            false, A, false, Bhi, (short)0, zeroC, false, false);

        // re-layout: each lane collects its pixel's 16 sigmas in depth order
        float sig[16];
        #pragma unroll
        for (int k = 0; k < 8; ++k) {
            float o0 = __shfl_xor(d0[k], 16, 32);
            float o1 = __shfl_xor(d1[k], 16, 32);
            sig[k]     = lo ? d0[k] : o1;
            sig[k + 8] = lo ? o0 : d1[k];
        }

        // front-to-back compositing over this chunk (fully unrolled:
        // keeps sig[] in named VGPRs, no v_movrels indexed access)
        #pragma unroll
        for (int n = 0; n < 16; ++n) {
            float s = sig[n];
            float4 cg = sRGBV[cb + n];              // LDS broadcast
            float alpha = fminf(__expf(-s), 0.999f);
            bool keep = (cg.w > 0.5f) && (s >= 0.0f) && (alpha >= (1.0f / 255.0f));
            float av = keep ? alpha : 0.0f;
            float w = T * av;
            accR += w * cg.x;
            accG += w * cg.y;
            accB += w * cg.z;
            T *= (1.0f - av);
        }
    }

    const int px = tx0 + lane;
    const int o = (row * IMG_W + px) * 3;
    out[o + 0] = fminf(fmaxf(accR + T * 0.149f,  0.0f), 1.0f);
    out[o + 1] = fminf(fmaxf(accG + T * 0.1647f, 0.0f), 1.0f);
    out[o + 2] = fminf(fmaxf(accB + T * 0.2157f, 0.0f), 1.0f);
}

// ---------------------------------------------------------------------------
extern "C" void kernel_launch(void* const* d_in, const int* in_sizes, int n_in,
                              void* d_out, int out_size, void* d_ws, size_t ws_size,
                              hipStream_t stream) {
    const float* points       = (const float*)d_in[0];
    const float* mean_params  = (const float*)d_in[1];
    const float* scale_params = (const float*)d_in[2];
    const float* quat_params  = (const float*)d_in[3];
    const float* colors       = (const float*)d_in[4];
    const float* shs          = (const float*)d_in[5];
    const float* viewmat      = (const float*)d_in[6];
    const float* Kmat         = (const float*)d_in[7];
    float* out = (float*)d_out;

    float* ws = (float*)d_ws;
    float*  zkey  = ws;                         // 1024
    float4* F0    = (float4*)(ws + 2048);       // 1024 * 4
    float*  CC    = ws + 6144;                  // 1024
    float4* RGBV  = (float4*)(ws + 7168);       // 1024 * 4
    float4* SF0   = (float4*)(ws + 11264);      // 1024 * 4
    float*  SCC   = ws + 15360;                 // 1024
    float4* SRGBV = (float4*)(ws + 16384);      // 1024 * 4

    preprocess_kernel<<<NPTS / 256, 256, 0, stream>>>(
        points, mean_params, scale_params, quat_params, colors, shs,
        viewmat, Kmat, zkey, F0, CC, RGBV);

    sort_kernel<<<1, NPTS, 0, stream>>>(zkey, F0, CC, RGBV, SF0, SCC, SRGBV);

    // 1536 tiles of 32 pixels, 8 waves (tiles) per 256-thread block
    raster_kernel<<<(IMG_H * IMG_W / 32) / 8, 256, 0, stream>>>(
        SF0, SCC, SRGBV, out);
}